// GCNEncoder_4604204941840
// MI455X (gfx1250) — compile-verified
//
#include <hip/hip_runtime.h>
#include <hip/hip_bf16.h>

// ---------------------------------------------------------------------------
// GCN encoder for MI455X (gfx1250, wave32).
// GEMMs: V_WMMA_F32_16X16X4_F32, double-buffered async LDS staging of A
// (global_load_async_to_lds_b128 / s_wait_asynccnt), and K-pair-repacked
// weights so every B fragment is one b64 load with an immediate offset.
// Edge aggregation: wave-per-edge float4 gathers (L2-resident features) +
// global f32 atomics. That phase is the HBM/L2 bandwidth bound.
// ---------------------------------------------------------------------------

#define N_NODES 100000
#define N_EDGES 3200000
#define IN_C    128
#define HID     256
#define OUT_C   128
#define N_GRAPHS 64
#define LN_EPS  1e-5f

typedef __attribute__((ext_vector_type(2))) float v2f;
typedef __attribute__((ext_vector_type(8))) float v8f;

// ---------------------------- utility kernels ------------------------------

__global__ void k_zero_f32(float* __restrict__ p, size_t n) {
    size_t i = (size_t)blockIdx.x * blockDim.x + threadIdx.x;
    size_t stride = (size_t)gridDim.x * blockDim.x;
    for (; i < n; i += stride) p[i] = 0.0f;
}

__global__ void k_deg_count(const int* __restrict__ dst, float* __restrict__ deg, int E) {
    int i = blockIdx.x * blockDim.x + threadIdx.x;
    int stride = gridDim.x * blockDim.x;
    for (; i < E; i += stride) atomicAdd(&deg[dst[i]], 1.0f);
}

__global__ void k_deg_finalize(float* __restrict__ deg, int n) {
    int i = blockIdx.x * blockDim.x + threadIdx.x;
    if (i < n) deg[i] = rsqrtf(deg[i] + 1.0f);   // +1 self loop; always > 0
}

// Repack W[K,N] -> Wp[K/2][N][2]: Wp2[kp*N + n] = { W[2kp][n], W[2kp+1][n] }.
// Makes the per-lane WMMA B fragment (two K-adjacent values of one column)
// a single 8-byte load.
__global__ void k_repack_w(const float* __restrict__ W, float* __restrict__ Wp,
                           int total /* (K/2)*N */, int nshift /* log2(N) */) {
    int i = blockIdx.x * blockDim.x + threadIdx.x;
    if (i < total) {
        int kp = i >> nshift;
        int n  = i & ((1 << nshift) - 1);
        int N  = 1 << nshift;
        float2 v;
        v.x = W[(2 * kp) * N + n];
        v.y = W[(2 * kp + 1) * N + n];
        ((float2*)Wp)[i] = v;
    }
}

// ------------------------------ WMMA GEMM ----------------------------------
// H[M,N] = A[M,K] @ W[K,N] with W pre-packed as above.
// Block = 256 threads (8 waves), computes a 32-row x N-col stripe:
//   wave w owns columns [w*16*TPW, (w+1)*16*TPW) and both 16-row groups, so
//   each B fragment feeds 2 WMMAs and each A fragment feeds TPW WMMAs.
// A stripe staged 32xKC per K-slab into double-buffered LDS with CDNA5 async
// global->LDS DMA, overlapped with compute. N is compile-time so all inner
// B/store offsets fold into instruction offset immediates.
// Requires: M % 32 == 0 (100000/32 = 3125), K % KC == 0.
template<int N>
__global__ __launch_bounds__(256)
void k_gemm_wmma_f32(const float* __restrict__ A,
                     const float* __restrict__ Wp,   // packed [K/2][N][2]
                     float* __restrict__ H,
                     int K) {
    constexpr int TPW = N / 128;           // 16-col tiles per wave
    constexpr int KC  = 64;                // K-slab per stage
    __shared__ float Asl[2][32 * KC];      // 2 x 8KB double buffer

    const int tid  = threadIdx.x;
    const int wave = tid >> 5;
    const int lane = tid & 31;
    const int m0   = blockIdx.x * 32;

    const int mrow = lane & 15;            // A lane row within 16-row group
    const int kgrp = (lane >> 4) << 1;     // lanes 16-31 hold K+2,K+3
    const int ncol = lane & 15;            // B/C/D lane column
    const int n0   = wave * (16 * TPW);

    // Staging map: 512 float4 chunks (32 rows x 16 chunks); 2 chunks/thread.
    const int srow0 = tid >> 4;                   // rows 0..15
    const int scol0 = (tid & 15) << 2;            // float offset within slab
    const int srow1 = srow0 + 16;                 // rows 16..31

    auto stage = [&](int buf, int kc) {
        const float* g0 = A + (m0 + srow0) * K + kc + scol0;
        const float* g1 = A + (m0 + srow1) * K + kc + scol0;
        unsigned l0 = (unsigned)(size_t)&Asl[buf][srow0 * KC + scol0];
        unsigned l1 = (unsigned)(size_t)&Asl[buf][srow1 * KC + scol0];
        // CDNA5 async DMA: global -> LDS, tracked by ASYNCcnt
        asm volatile("global_load_async_to_lds_b128 %0, %1, off"
                     :: "v"(l0), "v"(g0) : "memory");
        asm volatile("global_load_async_to_lds_b128 %0, %1, off"
                     :: "v"(l1), "v"(g1) : "memory");
    };

    v8f acc[2][TPW];
#pragma unroll
    for (int g = 0; g < 2; ++g)
#pragma unroll
        for (int t = 0; t < TPW; ++t) acc[g][t] = (v8f){};

    const float2* __restrict__ Wp2 = (const float2*)Wp;

    stage(0, 0);
    int buf = 0;
    for (int kc = 0; kc < K; kc += KC) {
        asm volatile("s_wait_asynccnt 0x0" ::: "memory");
        __syncthreads();                       // staged slab visible to all waves
        if (kc + KC < K) stage(buf ^ 1, kc + KC);

        const float* a0 = &Asl[buf][mrow * KC + kgrp];            // row group 0
        const float* a1 = a0 + 16 * KC;                           // row group 1
        // one base address; all k/t offsets below are compile-time immediates
        const float2* bbase = Wp2 + ((kc + kgrp) >> 1) * N + n0 + ncol;
#pragma unroll
        for (int k = 0; k < KC; k += 4) {
            v2f fa0 = *(const v2f*)(a0 + k);                      // ds_load_b64
            v2f fa1 = *(const v2f*)(a1 + k);
#pragma unroll
            for (int t = 0; t < TPW; ++t) {
                float2 w2 = bbase[(k >> 1) * N + t * 16];         // global_load_b64
                v2f fb;
                fb.x = w2.x;   // W[kc+k+kgrp  ][n]
                fb.y = w2.y;   // W[kc+k+kgrp+1][n]
                acc[0][t] = __builtin_amdgcn_wmma_f32_16x16x4_f32(
                    false, fa0, false, fb, (short)0, acc[0][t], false, false);
                acc[1][t] = __builtin_amdgcn_wmma_f32_16x16x4_f32(
                    false, fa1, false, fb, (short)0, acc[1][t], false, false);
            }
        }
        buf ^= 1;
    }

    // C/D layout: VGPR r holds row M = r + 8*(lane>>4), column = lane&15
    const int crow = 8 * (lane >> 4);
#pragma unroll
    for (int g = 0; g < 2; ++g) {
#pragma unroll
        for (int t = 0; t < TPW; ++t) {
            float* op = H + (m0 + g * 16 + crow) * N + n0 + t * 16 + ncol;
#pragma unroll
            for (int r = 0; r < 8; ++r) op[r * N] = acc[g][t][r];
        }
    }
}

// ------------------------- aggregation kernels -----------------------------

// agg[i,c] = h[i,c] * dinv[i]^2  (self-loop contribution; initializes agg)
__global__ void k_init_selfloop(const float* __restrict__ h,
                                const float* __restrict__ dinv,
                                float* __restrict__ agg,
                                size_t total, int cshift) {
    size_t i = (size_t)blockIdx.x * blockDim.x + threadIdx.x;
    size_t stride = (size_t)gridDim.x * blockDim.x;
    for (; i < total; i += stride) {
        float d = dinv[i >> cshift];
        agg[i] = h[i] * (d * d);
    }
}

// wave-per-edge scatter: agg[dst] += h[src] * dinv[src]*dinv[dst]
__global__ void k_scatter_edges(const int* __restrict__ src,
                                const int* __restrict__ dst,
                                const float* __restrict__ dinv,
                                const float* __restrict__ h,
                                float* __restrict__ agg,
                                int E, int C) {
    const int lane = threadIdx.x & 31;
    int wid = (blockIdx.x * blockDim.x + threadIdx.x) >> 5;
    const int nwaves = (gridDim.x * blockDim.x) >> 5;
    const int vecC = C >> 2;  // float4 elements per row

    for (int e = wid; e < E; e += nwaves) {
        const int s = src[e];
        const int d = dst[e];
        // prefetch next edge's source row while we chew on this one
        int en = e + nwaves;
        if (en < E) {
            const float* pf = h + (size_t)src[en] * C + 4 * lane;
            __builtin_prefetch(pf, 0, 0);
        }
        const float w = dinv[s] * dinv[d];
        const float4* __restrict__ hp = (const float4*)(h + (size_t)s * C);
        float* __restrict__ ap = agg + (size_t)d * C;
#pragma unroll 2
        for (int c = lane; c < vecC; c += 32) {
            float4 v = hp[c];
            int base = c << 2;
            atomicAdd(ap + base + 0, v.x * w);
            atomicAdd(ap + base + 1, v.y * w);
            atomicAdd(ap + base + 2, v.z * w);
            atomicAdd(ap + base + 3, v.w * w);
        }
    }
}

// a = leaky_relu(agg + bias); store back; accumulate per-column sum / sumsq
// blockDim.x == C (128 or 256); one column per thread, rows grid-strided.
__global__ void k_bias_act_stats(float* __restrict__ agg,
                                 const float* __restrict__ bias,
                                 float* __restrict__ colsum,
                                 float* __restrict__ colsumsq,
                                 int n, int C) {
    const int c = threadIdx.x;
    const float b = bias[c];
    float s = 0.0f, ss = 0.0f;
    for (int i = blockIdx.x; i < n; i += gridDim.x) {
        float v = agg[(size_t)i * C + c] + b;
        v = (v > 0.0f) ? v : 0.01f * v;   // leaky_relu, slope 0.01
        agg[(size_t)i * C + c] = v;
        s += v;
        ss += v * v;
    }
    atomicAdd(&colsum[c], s);
    atomicAdd(&colsumsq[c], ss);
}

__global__ void k_finalize_stats(const float* __restrict__ colsum,
                                 const float* __restrict__ colsumsq,
                                 const float* __restrict__ g,
                                 const float* __restrict__ be,
                                 float* __restrict__ scale,
                                 float* __restrict__ shift,
                                 int n, int C) {
    int c = blockIdx.x * blockDim.x + threadIdx.x;
    if (c < C) {
        float invn = 1.0f / (float)n;
        float mu = colsum[c] * invn;
        float var = colsumsq[c] * invn - mu * mu;
        float sc = rsqrtf(var + LN_EPS) * g[c];
        scale[c] = sc;
        shift[c] = be[c] - mu * sc;
    }
}

__global__ void k_normalize(float* __restrict__ agg,
                            const float* __restrict__ scale,
                            const float* __restrict__ shift,
                            size_t total, int cmask) {
    size_t i = (size_t)blockIdx.x * blockDim.x + threadIdx.x;
    size_t stride = (size_t)gridDim.x * blockDim.x;
    for (; i < total; i += stride) {
        int c = (int)(i & (size_t)cmask);
        agg[i] = agg[i] * scale[c] + shift[c];
    }
}

// ------------------------------- pooling -----------------------------------

// wave per node: out[batch[i]] += h[i]; cnt[batch[i]] += 1
__global__ void k_pool_scatter(const float* __restrict__ h,
                               const int* __restrict__ batch,
                               float* __restrict__ out,
                               float* __restrict__ cnt,
                               int n, int C) {
    const int lane = threadIdx.x & 31;
    int wid = (blockIdx.x * blockDim.x + threadIdx.x) >> 5;
    const int nwaves = (gridDim.x * blockDim.x) >> 5;
    const int vecC = C >> 2;

    for (int i = wid; i < n; i += nwaves) {
        const int b = batch[i];
        const float4* __restrict__ hp = (const float4*)(h + (size_t)i * C);
        float* __restrict__ op = out + (size_t)b * C;
        for (int c = lane; c < vecC; c += 32) {
            float4 v = hp[c];
            int base = c << 2;
            atomicAdd(op + base + 0, v.x);
            atomicAdd(op + base + 1, v.y);
            atomicAdd(op + base + 2, v.z);
            atomicAdd(op + base + 3, v.w);
        }
        if (lane == 0) atomicAdd(&cnt[b], 1.0f);
    }
}

__global__ void k_pool_div(float* __restrict__ out, const float* __restrict__ cnt,
                           int G, int C) {
    int i = blockIdx.x * blockDim.x + threadIdx.x;
    if (i < G * C) {
        float c = cnt[i / C];
        out[i] /= fmaxf(c, 1.0f);
    }
}

// ------------------------------- launcher ----------------------------------

extern "C" void kernel_launch(void* const* d_in, const int* in_sizes, int n_in,
                              void* d_out, int out_size, void* d_ws, size_t ws_size,
                              hipStream_t stream) {
    // Inputs (setup_inputs order):
    const float* x      = (const float*)d_in[0];   // [N, 128]
    const float* W_in   = (const float*)d_in[1];   // [128, 256]
    const float* b_in   = (const float*)d_in[2];
    const float* g_in   = (const float*)d_in[3];
    const float* be_in  = (const float*)d_in[4];
    const float* W_hid  = (const float*)d_in[5];   // [2, 256, 256]
    const float* b_hid  = (const float*)d_in[6];   // [2, 256]
    const float* g_hid  = (const float*)d_in[7];
    const float* be_hid = (const float*)d_in[8];
    const float* W_out  = (const float*)d_in[9];   // [256, 128]
    const float* b_out  = (const float*)d_in[10];
    const float* g_out  = (const float*)d_in[11];
    const float* be_out = (const float*)d_in[12];
    const int*   eidx   = (const int*)d_in[13];    // [2, E]
    const int*   batch  = (const int*)d_in[14];    // [N]
    float*       out    = (float*)d_out;           // [64, 128]

    const int* src = eidx;
    const int* dst = eidx + N_EDGES;

    // Workspace layout (fp32)
    float* bufH  = (float*)d_ws;                          // [N, 256] GEMM output
    float* bufA  = bufH + (size_t)N_NODES * HID;          // [N, 256] agg / activations
    float* dinv  = bufA + (size_t)N_NODES * HID;          // [N]
    float* csum  = dinv + N_NODES;                        // [256]
    float* csumq = csum + HID;                            // [256]
    float* scale = csumq + HID;                           // [256]
    float* shift = scale + HID;                           // [256]
    float* cnt   = shift + HID;                           // [64]
    float* wpack = cnt + N_GRAPHS;                        // [256*256] packed W

    // ---- degree / normalization coefficients ----
    k_zero_f32<<<256, 256, 0, stream>>>(dinv, (size_t)N_NODES);
    k_deg_count<<<(N_EDGES + 255) / 256, 256, 0, stream>>>(dst, dinv, N_EDGES);
    k_deg_finalize<<<(N_NODES + 255) / 256, 256, 0, stream>>>(dinv, N_NODES);

    // ---- layer descriptors ----
    struct Layer { const float *in, *W, *b, *g, *be; int K, C; };
    Layer layers[4] = {
        { x,    W_in,                b_in,        g_in,        be_in,        IN_C, HID   },
        { bufA, W_hid,               b_hid,       g_hid,       be_hid,       HID,  HID   },
        { bufA, W_hid + HID * HID,   b_hid + HID, g_hid + HID, be_hid + HID, HID,  HID   },
        { bufA, W_out,               b_out,       g_out,       be_out,       HID,  OUT_C },
    };

    for (int l = 0; l < 4; ++l) {
        const Layer& L = layers[l];
        const int C = L.C;
        const int cshift = (C == 256) ? 8 : 7;
        const size_t total = (size_t)N_NODES * C;

        // repack weights for b64 B-fragment loads
        const int wtotal = (L.K / 2) * C;
        k_repack_w<<<(wtotal + 255) / 256, 256, 0, stream>>>(L.W, wpack, wtotal, cshift);

        // GEMM: bufH = in @ W  (WMMA fp32, async-LDS double-buffered A)
        if (C == 256)
            k_gemm_wmma_f32<256><<<N_NODES / 32, 256, 0, stream>>>(L.in, wpack, bufH, L.K);
        else
            k_gemm_wmma_f32<128><<<N_NODES / 32, 256, 0, stream>>>(L.in, wpack, bufH, L.K);

        // agg = selfloop + edge scatter
        k_init_selfloop<<<2048, 256, 0, stream>>>(bufH, dinv, bufA, total, cshift);
        k_scatter_edges<<<8192, 256, 0, stream>>>(src, dst, dinv, bufH, bufA, N_EDGES, C);

        // bias + leaky_relu + column stats
        k_zero_f32<<<1, 256, 0, stream>>>(csum, (size_t)(2 * HID));  // csum+csumq contiguous
        k_bias_act_stats<<<1024, C, 0, stream>>>(bufA, L.b, csum, csumq, N_NODES, C);
        k_finalize_stats<<<1, C, 0, stream>>>(csum, csumq, L.g, L.be, scale, shift, N_NODES, C);
        k_normalize<<<2048, 256, 0, stream>>>(bufA, scale, shift, total, C - 1);
    }

    // ---- global mean pool ----
    k_zero_f32<<<32, 256, 0, stream>>>(out, (size_t)(N_GRAPHS * OUT_C));
    k_zero_f32<<<1, 64, 0, stream>>>(cnt, (size_t)N_GRAPHS);
    k_pool_scatter<<<4096, 256, 0, stream>>>(bufA, batch, out, cnt, N_NODES, OUT_C);
    k_pool_div<<<(N_GRAPHS * OUT_C + 255) / 256, 256, 0, stream>>>(out, cnt, N_GRAPHS, OUT_C);
}